// Detail_loss_309237645405
// MI455X (gfx1250) — compile-verified
//
#include <hip/hip_runtime.h>

// Problem geometry (from reference): (2,7,7,3,256,256) -> 98 images of 3x256x256.
#define W      256
#define H      256
#define NB     98            // 2*7*7 batches
#define RB     8             // output rows per workgroup
#define LR     (RB + 2)      // staged rows incl. +/-1 halo
#define NBX    (H / RB)      // 32 row-blocks per batch
#define NPART  (NBX * NB)    // 3136 per-workgroup partials

typedef float v2f __attribute__((ext_vector_type(2)));
typedef float v8f __attribute__((ext_vector_type(8)));

// Flat address of LDS object: high 32 bits = shared aperture, low 32 = LDS byte
// address (CDNA5 ISA 10.2). Truncate to get the DS address for async copies.
__device__ __forceinline__ unsigned lds_addr32(const void* p) {
    return (unsigned)(unsigned long long)p;
}

// CDNA5 async bulk copy: 16B per active lane, global -> LDS, tracked by ASYNCcnt.
__device__ __forceinline__ void async_ld16(unsigned lds_off, const float* g) {
    asm volatile("global_load_async_to_lds_b128 %0, %1, off"
                 :: "v"(lds_off), "v"((unsigned long long)g)
                 : "memory");
}

__device__ __forceinline__ void wait_async0() {
#if __has_builtin(__builtin_amdgcn_s_wait_asynccnt)
    __builtin_amdgcn_s_wait_asynccnt(0);
#else
    asm volatile("s_wait_asynccnt 0" ::: "memory");
#endif
}

// Exact f32 sum of the 32 lane values of the calling (fully active) wave.
// Uses V_WMMA_F32_16X16X4_F32 with A = per-lane partials (a.x=s, a.y=0) and
// B = all-ones: D[m][n] = s[m] + s[m+16] for every n, so the 8 accumulator
// VGPRs hold the 16 pair-sums; 7 adds + one xor-16 shuffle finish the sum.
__device__ __forceinline__ float wave_sum32(float s) {
#if __has_builtin(__builtin_amdgcn_wmma_f32_16x16x4_f32)
    v2f a;  a[0] = s;    a[1] = 0.0f;   // A(16x4): lanes0-15 -> K0/K1, lanes16-31 -> K2/K3
    v2f b;  b[0] = 1.0f; b[1] = 1.0f;   // B(4x16): all ones regardless of slot mapping
    v8f c = {0.f, 0.f, 0.f, 0.f, 0.f, 0.f, 0.f, 0.f};
    c = __builtin_amdgcn_wmma_f32_16x16x4_f32(
            /*neg_a=*/false, a, /*neg_b=*/false, b,
            /*c_mod=*/(short)0, c, /*reuse_a=*/false, /*reuse_b=*/false);
    float t = c[0] + c[1] + c[2] + c[3] + c[4] + c[5] + c[6] + c[7];
    t += __shfl_xor(t, 16, 32);         // lanes0-15 hold rows 0-7, lanes16-31 rows 8-15
    return t;
#else
    for (int off = 16; off > 0; off >>= 1) s += __shfl_xor(s, off, 32);
    return s;
#endif
}

__global__ __launch_bounds__(256)
void detail_loss_partials(const float* __restrict__ infer,
                          const float* __restrict__ ref,
                          float* __restrict__ partials) {
    const int tid = threadIdx.x;
    const int rblk = blockIdx.x;        // 0..31
    const int b    = blockIdx.y;        // 0..97
    const int r0   = rblk * RB;         // first output row of this block

    __shared__ __align__(16) float raw[2][LR][3][W];   // async-copy landing zone
    __shared__ float S[2][LR][W + 2];                  // channel-sum plane, padded cols
    __shared__ float red[256];

    // ---- Phase 1: async-copy LR rows x 3 channels x {infer,ref} into LDS ----
    // 2*LR*3*(W/4) = 3840 16B chunks = 15 per thread (EXEC-uniform loop).
    const int NCHUNK = 2 * LR * 3 * (W / 4);
    for (int q = tid; q < NCHUNK; q += 256) {
        const int t   = q / (LR * 3 * (W / 4));
        int qq        = q - t * (LR * 3 * (W / 4));
        const int r   = qq / (3 * (W / 4));
        qq           -= r * (3 * (W / 4));
        const int c   = qq / (W / 4);
        const int xq  = qq - c * (W / 4);
        int gr = r0 - 1 + r;
        gr = gr < 0 ? 0 : (gr > H - 1 ? H - 1 : gr);   // clamp; zeroed logically below
        const float* base = t ? ref : infer;
        const float* src  = base + ((size_t)(b * 3 + c) * H + (size_t)gr) * W + xq * 4;
        async_ld16(lds_addr32(&raw[t][r][c][xq * 4]), src);
    }
    wait_async0();
    __syncthreads();

    // ---- Phase 2: channel-sum plane with zero-padded columns / OOB rows ----
    for (int q = tid; q < 2 * LR * W; q += 256) {
        const int t = q / (LR * W);
        const int r = (q / W) % LR;
        const int x = q % W;
        const int gr = r0 - 1 + r;
        const bool ok = (gr >= 0) && (gr < H);
        const float v = ok ? (raw[t][r][0][x] + raw[t][r][1][x] + raw[t][r][2][x]) : 0.0f;
        S[t][r][x + 1] = v;
    }
    if (tid < 2 * LR) {
        const int t = tid / LR, r = tid % LR;
        S[t][r][0] = 0.0f;
        S[t][r][W + 1] = 0.0f;
    }
    __syncthreads();

    // ---- Phase 3: |delta grad_h| + |delta grad_v| over RB x W positions ----
    float acc = 0.0f;
    const float(*Si)[W + 2] = S[0];
    const float(*Sr)[W + 2] = S[1];
    for (int q = tid; q < RB * W; q += 256) {
        const int r  = q / W;       // 0..RB-1 -> global row r0+r
        const int x  = q % W;
        const int lr = r + 1;       // S row for this output row
        const int xp = x + 1;       // padded column index
        const float gh = 0.5f * ((Si[lr][xp + 1] - Si[lr][xp - 1]) -
                                 (Sr[lr][xp + 1] - Sr[lr][xp - 1]));
        const float gv = 0.5f * ((Si[lr + 1][xp] - Si[lr - 1][xp]) -
                                 (Sr[lr + 1][xp] - Sr[lr - 1][xp]));
        acc += fabsf(gh) + fabsf(gv);
    }

    // ---- Phase 4: block reduce (WMMA wave reduction), one partial per WG ----
    red[tid] = acc;
    __syncthreads();
    if (tid < 32) {
        float s = 0.0f;
#pragma unroll
        for (int i = 0; i < 8; ++i) s += red[tid + 32 * i];
        const float total = wave_sum32(s);
        if (tid == 0) partials[b * NBX + rblk] = total;
    }
}

__global__ __launch_bounds__(256)
void detail_loss_finalize(const float* __restrict__ partials,
                          float* __restrict__ out) {
    __shared__ float red[256];
    const int tid = threadIdx.x;
    float acc = 0.0f;
    for (int i = tid; i < NPART; i += 256) acc += partials[i];
    red[tid] = acc;
    __syncthreads();
    if (tid < 32) {
        float s = 0.0f;
#pragma unroll
        for (int i = 0; i < 8; ++i) s += red[tid + 32 * i];
        const float total = wave_sum32(s);
        if (tid == 0) {
            // (loss_h + loss_v)/2 = (sum_h + sum_v) / (2 * 98 * 258 * 256)
            const float scale = 1.0f / (2.0f * 98.0f * 258.0f * 256.0f);
            out[0] = total * scale;
        }
    }
}

extern "C" void kernel_launch(void* const* d_in, const int* in_sizes, int n_in,
                              void* d_out, int out_size, void* d_ws, size_t ws_size,
                              hipStream_t stream) {
    (void)in_sizes; (void)n_in; (void)out_size; (void)ws_size;
    const float* infer = (const float*)d_in[0];
    const float* ref   = (const float*)d_in[1];
    float* partials    = (float*)d_ws;       // NPART floats of scratch
    float* out         = (float*)d_out;

    dim3 grid(NBX, NB);
    detail_loss_partials<<<grid, 256, 0, stream>>>(infer, ref, partials);
    detail_loss_finalize<<<1, 256, 0, stream>>>(partials, out);
}